// ImageTreensformerV6_32255204393441
// MI455X (gfx1250) — compile-verified
//
#include <hip/hip_runtime.h>

typedef __attribute__((ext_vector_type(2))) float v2f;
typedef __attribute__((ext_vector_type(4))) float v4f;
typedef __attribute__((ext_vector_type(8))) float v8f;

// ---------------------------------------------------------------------------
// Phase 1a: per-plane partial reduction with b128 loads.
// grid = 768 blocks (plane = b*3+c in [0,192), slice s in [0,4)), 256 threads.
// Thread t: q = bh-within-slice (t>>6), pi = (t>>2)&15, pjg = t&3.
// Loads float4 x[plane, (s*4+q)*16+pi, bw*16 + pjg*4] over bw in [0,16):
//   16 global_load_b128 per thread, 64B-contiguous per 4-lane group.
// q-slices combined deterministically via LDS -> partials[blk*256 + pi*16+pj].
// ---------------------------------------------------------------------------
__global__ void reduce_partial(const float* __restrict__ x, float* __restrict__ partials) {
    __shared__ float lds[1024];
    const int blk   = blockIdx.x;        // plane*4 + s
    const int plane = blk >> 2;          // b*3 + c
    const int s     = blk & 3;
    const int t     = threadIdx.x;       // 0..255
    const int q     = t >> 6;            // bh within slice, 0..3
    const int pi    = (t >> 2) & 15;     // intra-patch row
    const int pjg   = t & 3;             // pj group of 4

    const float* row = x + (size_t)plane * 65536
                     + (size_t)(((s * 4 + q) * 16 + pi) * 256) + pjg * 4;
    v4f sum = {0.f, 0.f, 0.f, 0.f};
#pragma unroll
    for (int bw = 0; bw < 16; ++bw)
        sum += *(const v4f*)(row + bw * 16);

    // element (q, pi, pj=pjg*4+c) lands at lds[q*256 + pi*16 + pj]
    *(v4f*)&lds[t * 4] = sum;
    __syncthreads();

    // output index t -> (pi2 = t>>4, pj2 = t&15); sum q slices in fixed order
    float r = lds[t] + lds[256 + t] + lds[512 + t] + lds[768 + t];
    partials[blk * 256 + t] = r;
}

// ---------------------------------------------------------------------------
// Phase 1b: deterministic combine of the 4 s-slices -> m[b*768 + f] (mean).
// grid = 192 blocks, 256 threads; idx = b*768 + c*256 + pi*16 + pj.
// ---------------------------------------------------------------------------
__global__ void reduce_combine(const float* __restrict__ partials, float* __restrict__ m) {
    const int idx    = blockIdx.x * 256 + threadIdx.x;
    const int plane  = idx >> 8;
    const int within = idx & 255;
    const float* p = partials + plane * 1024 + within;
    m[idx] = (p[0] + p[256] + p[512] + p[768]) * (1.0f / 256.0f);
}

// ---------------------------------------------------------------------------
// Phase 2: root (64x128) = m (64x768) @ W_emb^T + b_emb + pos4, via WMMA f32.
// grid = (4 M-tiles, 8 N-tiles), 32 threads = one wave per 16x16 tile.
// A frag: lane%16 = M row, K = (lane/16)*2 + v ; B frag: lane%16 = N col, same K.
// ---------------------------------------------------------------------------
__global__ void gemm_root(const float* __restrict__ m, const float* __restrict__ W_emb,
                          const float* __restrict__ b_emb, const float* __restrict__ pos4,
                          float* __restrict__ root) {
    const int tm   = blockIdx.x;            // 0..3   (batch tile)
    const int tn   = blockIdx.y;            // 0..7   (feature tile)
    const int lane = threadIdx.x;
    const int l16  = lane & 15;
    const int half = lane >> 4;
    const int rowA = tm * 16 + l16;         // batch row
    const int d    = tn * 16 + l16;         // output feature
    const float* aptr = m     + rowA * 768 + half * 2;
    const float* bptr = W_emb + d    * 768 + half * 2;
    v8f acc = {};
    for (int k = 0; k < 768; k += 4) {
        v2f a = *(const v2f*)(aptr + k);
        v2f b = *(const v2f*)(bptr + k);
        acc = __builtin_amdgcn_wmma_f32_16x16x4_f32(false, a, false, b,
                                                    (short)0, acc, false, false);
    }
    const float bias = b_emb[d] + pos4[d];
#pragma unroll
    for (int i = 0; i < 8; ++i) {
        const int brow = tm * 16 + half * 8 + i;   // C/D layout: M = half*8 + vgpr
        root[brow * 128 + d] = acc[i] + bias;
    }
}

// ---------------------------------------------------------------------------
// Phase 3: logits (64x1000) = root (64x128) @ W_cls^T + b_cls, via WMMA f32.
// grid = (4 M-tiles, 63 N-tiles); N tail (1000..1007) computed as zeros, not stored.
// ---------------------------------------------------------------------------
__global__ void gemm_logits(const float* __restrict__ root, const float* __restrict__ W_cls,
                            const float* __restrict__ b_cls, float* __restrict__ out) {
    const int tm   = blockIdx.x;            // 0..3
    const int tn   = blockIdx.y;            // 0..62
    const int lane = threadIdx.x;
    const int l16  = lane & 15;
    const int half = lane >> 4;
    const int rowA = tm * 16 + l16;
    const int n    = tn * 16 + l16;         // class index
    const bool nvalid = (n < 1000);
    const float* aptr = root + rowA * 128 + half * 2;
    const float* bptr = W_cls + (nvalid ? n : 0) * 128 + half * 2;  // clamp OOB rows
    v8f acc = {};
#pragma unroll
    for (int k = 0; k < 128; k += 4) {
        v2f a = *(const v2f*)(aptr + k);
        v2f b = *(const v2f*)(bptr + k);
        if (!nvalid) { b[0] = 0.0f; b[1] = 0.0f; }
        acc = __builtin_amdgcn_wmma_f32_16x16x4_f32(false, a, false, b,
                                                    (short)0, acc, false, false);
    }
    if (nvalid) {
        const float bias = b_cls[n];
#pragma unroll
        for (int i = 0; i < 8; ++i) {
            const int brow = tm * 16 + half * 8 + i;
            out[brow * 1000 + n] = acc[i] + bias;
        }
    }
}

// ---------------------------------------------------------------------------
// Inputs (setup_inputs order): 0:x 1:W_emb 2:b_emb 3:pos0 4:pos1 5:pos2 6:pos3
//                              7:pos4 8:W_cls 9:b_cls. Only level-4 survives.
// ---------------------------------------------------------------------------
extern "C" void kernel_launch(void* const* d_in, const int* in_sizes, int n_in,
                              void* d_out, int out_size, void* d_ws, size_t ws_size,
                              hipStream_t stream) {
    const float* x     = (const float*)d_in[0];
    const float* W_emb = (const float*)d_in[1];
    const float* b_emb = (const float*)d_in[2];
    const float* pos4  = (const float*)d_in[7];
    const float* W_cls = (const float*)d_in[8];
    const float* b_cls = (const float*)d_in[9];

    float* ws       = (float*)d_ws;
    float* m        = ws;                       // 64*768      = 49152 floats
    float* partials = ws + 49152;               // 192*4*256   = 196608 floats
    float* root     = ws + 49152 + 196608;      // 64*128      = 8192 floats

    reduce_partial <<<768, 256, 0, stream>>>(x, partials);
    reduce_combine <<<192, 256, 0, stream>>>(partials, m);
    gemm_root      <<<dim3(4, 8),  32, 0, stream>>>(m, W_emb, b_emb, pos4, root);
    gemm_logits    <<<dim3(4, 63), 32, 0, stream>>>(root, W_cls, b_cls, (float*)d_out);
}